// SGC_4698694222239
// MI455X (gfx1250) — compile-verified
//
#include <hip/hip_runtime.h>
#include <stdint.h>

#define N_NODES 100000
#define N_EDGES 1600000
#define D_FEAT  128

#define TILE   2048           // edges per LDS tile
#define BLOCK  256            // 8 waves (wave32)
#define WAVES  (BLOCK / 32)
#define EDGES_PER_WAVE (TILE / WAVES)   // 256
#define ASYNC_OPS_PER_TILE 6            // 3 regions * 2 b128 per thread

// Low 32 bits of a generic pointer into LDS == LDS byte offset.
__device__ __forceinline__ uint32_t lds_off(const void* p) {
    return (uint32_t)(uintptr_t)p;
}

// CDNA5 async copy: global -> LDS, 16 bytes per active lane, tracked by ASYNCcnt.
__device__ __forceinline__ void async_copy_b128(uint32_t lds_addr, const void* gaddr) {
    asm volatile("global_load_async_to_lds_b128 %0, %1, off"
                 :: "v"(lds_addr), "v"(gaddr) : "memory");
}

// ---------------------------------------------------------------------------
// Kernel (emitted first so the disasm snippet shows it):
//   out[dst] += (1-alpha) * w_e * x[src]   via LDS-staged edge tiles.
// One wave32 handles a full 128-float row per edge (float4 per lane).
// Edge stream double-buffered into LDS with ASYNCcnt-tracked b128 copies.
// ---------------------------------------------------------------------------
__global__ void __launch_bounds__(BLOCK)
sgc_scatter_kernel(const float* __restrict__ x, const int* __restrict__ ei,
                   const float* __restrict__ ew, const float* __restrict__ alpha,
                   float* __restrict__ out) {
    __shared__ __align__(16) int   s_dst[2][TILE];
    __shared__ __align__(16) int   s_src[2][TILE];
    __shared__ __align__(16) float s_w[2][TILE];

    const int t    = threadIdx.x;
    const int wave = t >> 5;
    const int lane = t & 31;
    const float scale = 1.0f - alpha[0];

    const int* __restrict__ dstp = ei;            // edge_index[0] (row / dst)
    const int* __restrict__ srcp = ei + N_EDGES;  // edge_index[1] (col / src)

    const int ntiles = (N_EDGES + TILE - 1) / TILE;   // 782
    const int stride = gridDim.x;

    // Stage one tile (3 regions x 2048 words) into LDS buffer `buf`.
    // Every thread issues exactly ASYNC_OPS_PER_TILE b128 async ops (tail
    // clamped so the per-thread ASYNCcnt stays uniform across the block).
    auto stage = [&](int tile, int buf) {
        const long base = (long)tile * TILE;
        #pragma unroll
        for (int k = 0; k < 2; ++k) {
            const int  slot = k * BLOCK + t;          // float4 slot 0..511
            long wi = base + (long)slot * 4;          // element index of this float4
            if (wi > (long)N_EDGES - 4) wi = (long)N_EDGES - 4;  // clamp tail
            const uint32_t lo = (uint32_t)slot * 16u;
            async_copy_b128(lds_off(&s_dst[buf][0]) + lo, dstp + wi);
            async_copy_b128(lds_off(&s_src[buf][0]) + lo, srcp + wi);
            async_copy_b128(lds_off(&s_w[buf][0])   + lo, ew   + wi);
        }
    };

    int tile = blockIdx.x;
    if (tile >= ntiles) return;

    stage(tile, 0);
    int buf = 0;

    for (; tile < ntiles; tile += stride) {
        const int  nxt = tile + stride;
        const bool pre = (nxt < ntiles);
        if (pre) {
            stage(nxt, buf ^ 1);                      // overlap: next tile in flight
            asm volatile("s_wait_asynccnt 6" ::: "memory");  // current tile done
        } else {
            asm volatile("s_wait_asynccnt 0" ::: "memory");
        }
        __syncthreads();

        const long ebase = (long)tile * TILE;
        const int  e0    = wave * EDGES_PER_WAVE;
        #pragma unroll 2
        for (int e = e0; e < e0 + EDGES_PER_WAVE; ++e) {
            if (ebase + e >= N_EDGES) break;          // only last tile is partial
            const int   d   = s_dst[buf][e];          // LDS broadcast reads
            const int   s   = s_src[buf][e];
            const float wgt = s_w[buf][e] * scale;

            // Pull next edge's source row toward the WGP while we work.
            if (e + 1 < e0 + EDGES_PER_WAVE) {
                const int sn = s_src[buf][e + 1];
                __builtin_prefetch(x + (long)sn * D_FEAT + lane * 4, 0, 0);  // global_prefetch_b8
            }

            const float4 xv = ((const float4*)(x + (long)s * D_FEAT))[lane];  // global_load_b128
            float* op = out + (long)d * D_FEAT + lane * 4;
            unsafeAtomicAdd(op + 0, xv.x * wgt);      // global_atomic_add_f32 (no return)
            unsafeAtomicAdd(op + 1, xv.y * wgt);
            unsafeAtomicAdd(op + 2, xv.z * wgt);
            unsafeAtomicAdd(op + 3, xv.w * wgt);
        }
        __syncthreads();
        buf ^= 1;
    }
}

// ---------------------------------------------------------------------------
// Kernel: out = alpha * x   (also re-initializes poisoned d_out every call)
// ---------------------------------------------------------------------------
__global__ void __launch_bounds__(BLOCK)
sgc_scale_kernel(const float* __restrict__ x, const float* __restrict__ alpha,
                 float* __restrict__ out) {
    const long n4 = (long)N_NODES * D_FEAT / 4;     // 3,200,000 float4s
    long i = (long)blockIdx.x * blockDim.x + threadIdx.x;
    if (i < n4) {
        float a = alpha[0];
        float4 v = ((const float4*)x)[i];
        float4 r;
        r.x = a * v.x; r.y = a * v.y; r.z = a * v.z; r.w = a * v.w;
        ((float4*)out)[i] = r;
    }
}

// ---------------------------------------------------------------------------
extern "C" void kernel_launch(void* const* d_in, const int* in_sizes, int n_in,
                              void* d_out, int out_size, void* d_ws, size_t ws_size,
                              hipStream_t stream) {
    const float* x     = (const float*)d_in[0];   // [N_NODES, D_FEAT] f32
    const int*   ei    = (const int*)d_in[1];     // [2, N_EDGES] i32
    const float* ew    = (const float*)d_in[2];   // [N_EDGES] f32
    const float* alpha = (const float*)d_in[3];   // [1] f32
    float*       out   = (float*)d_out;           // [N_NODES, D_FEAT] f32

    // 1) out = alpha * x
    const int n4     = N_NODES * D_FEAT / 4;      // 3,200,000
    const int blocks = (n4 + BLOCK - 1) / BLOCK;  // 12,500
    sgc_scale_kernel<<<blocks, BLOCK, 0, stream>>>(x, alpha, out);

    // 2) scatter-add messages; 384 persistent blocks -> each runs 2-3 tiles,
    //    exercising the async double-buffer path.
    sgc_scatter_kernel<<<384, BLOCK, 0, stream>>>(x, ei, ew, alpha, out);
}